// PoolingLayer_69380901700064
// MI455X (gfx1250) — compile-verified
//
#include <hip/hip_runtime.h>
#include <math.h>

// Problem constants (match reference file).
#define NROWS 262144
#define DDIM  512
#define KTOP  65536

typedef float v2f __attribute__((ext_vector_type(2)));
typedef float v4f __attribute__((ext_vector_type(4)));
typedef float v8f __attribute__((ext_vector_type(8)));

// ---------------------------------------------------------------------------
// Kernel 0: inverse norm of p (512 elements), one block.
// ---------------------------------------------------------------------------
__global__ void pnorm_kernel(const float* __restrict__ p, float* __restrict__ inv_len) {
  __shared__ float s[256];
  int t = threadIdx.x;
  float a = p[t], b = p[t + 256];
  s[t] = a * a + b * b;
  __syncthreads();
  for (int off = 128; off > 0; off >>= 1) {
    if (t < off) s[t] += s[t + off];
    __syncthreads();
  }
  if (t == 0) *inv_len = 1.0f / sqrtf(s[0]);
}

// ---------------------------------------------------------------------------
// Kernel 1: scores[r] = dot(x[r,:], p) via V_WMMA_F32_16X16X4_F32.
// One wave32 owns a 16-row tile. Lane layout for f32 A (16x4):
//   lanes 0-15 hold (M=lane, K slots 0,1), lanes 16-31 hold (M=lane-16, K slots 2,3).
// A single float4 load per lane feeds two WMMAs (K-chunk of 8); the
// slot->memory mapping is chosen consistently for A and B. B has p broadcast
// to all 16 columns, so C holds the score replicated across columns.
//
// p (2 KB) is staged into LDS once per block via GLOBAL_LOAD_ASYNC_TO_LDS_B128
// (ASYNCcnt-tracked async copy), so the hot loop only streams x from HBM.
// ---------------------------------------------------------------------------
__global__ void matvec_wmma_kernel(const float* __restrict__ x,
                                   const float* __restrict__ p,
                                   float* __restrict__ scores) {
  __shared__ __align__(16) float plds[DDIM];   // 2 KB

  const int lane = threadIdx.x & 31;
  const int wave = threadIdx.x >> 5;
  const int rowBase = (blockIdx.x * 8 + wave) * 16;
  const int half = lane >> 4;          // 0: K slots 0,1   1: K slots 2,3
  const int r = rowBase + (lane & 15);
  const float* xrow = x + (size_t)r * DDIM + half * 4;

  // Async-stage p into LDS: wave 0 issues 4 copies of 512 B (16 B / lane).
  if (wave == 0) {
    unsigned loff = (unsigned)(size_t)(&plds[0]) + (unsigned)(lane * 16);
    unsigned long long gaddr =
        (unsigned long long)(size_t)p + (unsigned long long)(lane * 16);
    #pragma unroll
    for (int c = 0; c < 4; c++) {
      asm volatile("global_load_async_to_lds_b128 %0, %1, off"
                   :: "v"(loff + (unsigned)(c * 512)),
                      "v"(gaddr + (unsigned long long)(c * 512))
                   : "memory");
    }
    asm volatile("s_wait_asynccnt 0x0" ::: "memory");
  }
  __syncthreads();

  const float* pl = plds + half * 4;

#if __has_builtin(__builtin_amdgcn_wmma_f32_16x16x4_f32)
  v8f c = {};
  #pragma unroll 4
  for (int k = 0; k < DDIM; k += 8) {
    v4f a4 = *(const v4f*)(xrow + k);
    v4f b4 = *(const v4f*)(pl + k);     // ds_load_b128 broadcast
    v2f a0 = {a4.x, a4.y}, a1 = {a4.z, a4.w};
    v2f b0 = {b4.x, b4.y}, b1 = {b4.z, b4.w};
    c = __builtin_amdgcn_wmma_f32_16x16x4_f32(false, a0, false, b0, (short)0, c,
                                              false, false);
    c = __builtin_amdgcn_wmma_f32_16x16x4_f32(false, a1, false, b1, (short)0, c,
                                              false, false);
  }
  // C layout: VGPR j, lanes 0-15 -> M=j, lanes 16-31 -> M=8+j (all columns equal).
  if ((lane & 15) == 0) {
    float* dst = scores + rowBase + half * 8;
    #pragma unroll
    for (int j = 0; j < 8; j++) dst[j] = c[j];
  }
#else
  // Fallback: plain FMA dot (keeps compile green; histogram will show wmma=0).
  v4f acc = {0.f, 0.f, 0.f, 0.f};
  for (int k = 0; k < DDIM; k += 8) {
    v4f a4 = *(const v4f*)(xrow + k);
    v4f b4 = *(const v4f*)(pl + k);
    acc += a4 * b4;
  }
  float sum = acc.x + acc.y + acc.z + acc.w;
  sum += __shfl_xor(sum, 16, 32);      // combine the two K-halves of a row
  if (lane < 16) scores[r] = sum;
#endif
}

// ---------------------------------------------------------------------------
// Kernel 2: build radix keys. Ascending sort of key == descending score.
// ---------------------------------------------------------------------------
__global__ void build_keys_kernel(const float* __restrict__ scores,
                                  unsigned* __restrict__ keys,
                                  unsigned* __restrict__ idx) {
  int i = blockIdx.x * blockDim.x + threadIdx.x;
  unsigned u = __float_as_uint(scores[i]);
  u ^= (u >> 31) ? 0xFFFFFFFFu : 0x80000000u;  // order-preserving float->uint
  keys[i] = ~u;                                 // invert for descending
  idx[i] = (unsigned)i;
}

// ---------------------------------------------------------------------------
// LSD radix sort: 8 passes x 4 bits, 64 tiles of 4096 (256 thr x 16 items).
// ---------------------------------------------------------------------------
#define RBITS   4
#define RADIX   16
#define RTHR    256
#define RITEMS  16
#define RTILE   (RTHR * RITEMS)     // 4096
#define RBLK    (NROWS / RTILE)     // 64

__global__ void radix_hist_kernel(const unsigned* __restrict__ keys,
                                  unsigned* __restrict__ hist, int shift) {
  __shared__ unsigned h[RADIX];
  int t = threadIdx.x;
  if (t < RADIX) h[t] = 0;
  __syncthreads();
  unsigned base = blockIdx.x * RTILE + t * RITEMS;
  #pragma unroll
  for (int i = 0; i < RITEMS; i++) {
    unsigned d = (keys[base + i] >> shift) & (RADIX - 1);
    atomicAdd(&h[d], 1u);
  }
  __syncthreads();
  if (t < RADIX) hist[t * RBLK + blockIdx.x] = h[t];  // digit-major layout
}

// Exclusive scan of RADIX*RBLK = 1024 entries, one block of 256 threads.
__global__ void radix_scan_kernel(unsigned* __restrict__ hist) {
  __shared__ unsigned part[RTHR];
  int t = threadIdx.x;
  unsigned v[4];
  unsigned sum = 0;
  #pragma unroll
  for (int i = 0; i < 4; i++) {
    v[i] = hist[t * 4 + i];
    unsigned tmp = sum; sum += v[i]; v[i] = tmp;   // local exclusive
  }
  part[t] = sum;
  __syncthreads();
  for (int off = 1; off < RTHR; off <<= 1) {       // Hillis-Steele inclusive
    unsigned xv = (t >= off) ? part[t - off] : 0u;
    __syncthreads();
    part[t] += xv;
    __syncthreads();
  }
  unsigned excl = (t == 0) ? 0u : part[t - 1];
  #pragma unroll
  for (int i = 0; i < 4; i++) hist[t * 4 + i] = v[i] + excl;
}

// Stable scatter: per-thread digit counters in LDS (16 digits x 256 threads),
// scanned digit-major so within-tile rank preserves memory order.
__global__ void radix_scatter_kernel(const unsigned* __restrict__ keysIn,
                                     const unsigned* __restrict__ idxIn,
                                     unsigned* __restrict__ keysOut,
                                     unsigned* __restrict__ idxOut,
                                     const unsigned* __restrict__ hist,
                                     int shift) {
  __shared__ unsigned cnt[RADIX][RTHR];   // 16 KB
  __shared__ unsigned part[RTHR];
  int t = threadIdx.x, blk = blockIdx.x;
  #pragma unroll
  for (int d = 0; d < RADIX; d++) cnt[d][t] = 0;
  __syncthreads();

  unsigned base = blk * RTILE + t * RITEMS;
  unsigned k[RITEMS], id[RITEMS], dg[RITEMS], lr[RITEMS];
  #pragma unroll
  for (int i = 0; i < RITEMS; i++) {
    k[i]  = keysIn[base + i];
    id[i] = idxIn[base + i];
    dg[i] = (k[i] >> shift) & (RADIX - 1);
    lr[i] = cnt[dg[i]][t]++;              // private column: no races
  }
  __syncthreads();

  // Exclusive scan of the 4096 counters, flattened digit-major (d, t).
  int d0 = t >> 4;
  int c0 = (t & 15) << 4;
  unsigned loc[16];
  unsigned run = 0;
  #pragma unroll
  for (int j = 0; j < 16; j++) { unsigned c = cnt[d0][c0 + j]; loc[j] = run; run += c; }
  part[t] = run;
  __syncthreads();
  for (int off = 1; off < RTHR; off <<= 1) {
    unsigned xv = (t >= off) ? part[t - off] : 0u;
    __syncthreads();
    part[t] += xv;
    __syncthreads();
  }
  unsigned excl = (t == 0) ? 0u : part[t - 1];
  #pragma unroll
  for (int j = 0; j < 16; j++) cnt[d0][c0 + j] = loc[j] + excl;
  __syncthreads();

  #pragma unroll
  for (int i = 0; i < RITEMS; i++) {
    unsigned d = dg[i];
    unsigned intile = cnt[d][t] - cnt[d][0] + lr[i];  // stable rank within digit
    unsigned pos = hist[d * RBLK + blk] + intile;
    keysOut[pos] = k[i];
    idxOut[pos]  = id[i];
  }
}

// ---------------------------------------------------------------------------
// Kernel 3: out[i,:] = x[top_idx[i],:] * (scores[top_idx[i]] * inv_len)
// One block per output row; 128 threads x float4 = 512 floats, coalesced.
// ---------------------------------------------------------------------------
__global__ void gather_scale_kernel(const float* __restrict__ x,
                                    const unsigned* __restrict__ sidx,
                                    const float* __restrict__ scores,
                                    const float* __restrict__ inv_len,
                                    float* __restrict__ out) {
  int row = blockIdx.x;
  unsigned src = sidx[row];
  float val = scores[src] * (*inv_len);
  const v4f* xs = (const v4f*)(x + (size_t)src * DDIM);
  v4f* os = (v4f*)(out + (size_t)row * DDIM);
  os[threadIdx.x] = xs[threadIdx.x] * val;
}

// ---------------------------------------------------------------------------
extern "C" void kernel_launch(void* const* d_in, const int* in_sizes, int n_in,
                              void* d_out, int out_size, void* d_ws, size_t ws_size,
                              hipStream_t stream) {
  const float* x = (const float*)d_in[0];
  const float* p = (const float*)d_in[1];
  float* out = (float*)d_out;

  // Workspace layout (256B aligned blocks): ~5.25 MiB total.
  char* ws = (char*)d_ws;
  const size_t N4 = (size_t)NROWS * 4;
  float*    inv_len = (float*)ws;
  float*    scores  = (float*)(ws + 256);
  unsigned* keysA   = (unsigned*)(ws + 256 + N4);
  unsigned* keysB   = (unsigned*)(ws + 256 + 2 * N4);
  unsigned* idxA    = (unsigned*)(ws + 256 + 3 * N4);
  unsigned* idxB    = (unsigned*)(ws + 256 + 4 * N4);
  unsigned* hist    = (unsigned*)(ws + 256 + 5 * N4);

  pnorm_kernel<<<1, 256, 0, stream>>>(p, inv_len);
  matvec_wmma_kernel<<<NROWS / 128, 256, 0, stream>>>(x, p, scores);
  build_keys_kernel<<<NROWS / 256, 256, 0, stream>>>(scores, keysA, idxA);

  unsigned *ki = keysA, *ko = keysB, *ii = idxA, *io = idxB;
  for (int pass = 0; pass < 8; pass++) {
    int shift = pass * RBITS;
    radix_hist_kernel<<<RBLK, RTHR, 0, stream>>>(ki, hist, shift);
    radix_scan_kernel<<<1, RTHR, 0, stream>>>(hist);
    radix_scatter_kernel<<<RBLK, RTHR, 0, stream>>>(ki, ii, ko, io, hist, shift);
    unsigned* tk = ki; ki = ko; ko = tk;
    unsigned* ti = ii; ii = io; io = ti;
  }
  // 8 passes (even) -> sorted result back in keysA/idxA (== ki/ii).
  gather_scale_kernel<<<KTOP, DDIM / 4, 0, stream>>>(x, ii, scores, inv_len, out);
}